// MultiScaleDeformableAttention_46342697124295
// MI455X (gfx1250) — compile-verified
//
#include <hip/hip_runtime.h>
#include <cmath>

typedef __attribute__((ext_vector_type(2))) float v2f;
typedef __attribute__((ext_vector_type(8))) float v8f;

#define B_    4
#define Q_    2048
#define D_    256
#define NH_   8
#define NL_   3
#define NPT_  4
#define HD_   32
#define ROWS_ (B_ * Q_)               /* 8192 */
#define LOFF_ (NH_ * NL_ * NPT_ * 2) /* 192  */
#define LATT_ (NH_ * NL_ * NPT_)     /* 96   */
#define MT_   (ROWS_ / 16)           /* 512 M-tiles */
#define MGRP_ (MT_ / 16)             /* 32 M-groups: 8 waves x 2 tiles per block */

// ---------------------------------------------------------------------------
// Shared GEMM body: block = one 16-col N-slice of B staged in LDS (pre-swizzled
// to float2 K-pairs so each lane's B operand is one ds_load_b64), 8 waves x
// 2 M-tiles. Per K-step: 2 global b64 (A) + 1 ds b64 (B) -> 2 WMMAs.
// ---------------------------------------------------------------------------
__device__ __forceinline__ void stage_B_lds(v2f* __restrict__ Bs,
                                            const float* __restrict__ Bm,
                                            int ldb, int col0, int tid) {
  const int n  = tid & 15;
  const int k0 = (tid >> 4) * 8;      // 8 K-pairs per thread: 16 threads/col-row group
#pragma unroll
  for (int i = 0; i < 8; ++i) {
    const int kp = k0 + i;            // K-pair index 0..127
    v2f t;
    t.x = Bm[(2 * kp)     * ldb + col0 + n];
    t.y = Bm[(2 * kp + 1) * ldb + col0 + n];
    Bs[kp * 16 + n] = t;
  }
}

__device__ __forceinline__ void wmma_dual_tile(const float* __restrict__ A0,
                                               const float* __restrict__ A1,
                                               const v2f* __restrict__ Bs,
                                               int lane, v8f& acc0, v8f& acc1) {
  const int mn = lane & 15;
  const int kh = (lane >> 4) << 1;    // 0 or 2
  const float* a0p = A0 + mn * D_ + kh;
  const float* a1p = A1 + mn * D_ + kh;
  const v2f*   bp  = Bs + ((kh >> 1) * 16 + mn);
#pragma unroll 8
  for (int k = 0; k < D_; k += 4) {
    v2f b  = bp[(k >> 1) * 16];                 // ds_load_b64
    v2f a0 = *(const v2f*)(a0p + k);            // global_load_b64
    v2f a1 = *(const v2f*)(a1p + k);            // global_load_b64
    acc0 = __builtin_amdgcn_wmma_f32_16x16x4_f32(false, a0, false, b,
                                                 (short)0, acc0, false, false);
    acc1 = __builtin_amdgcn_wmma_f32_16x16x4_f32(false, a1, false, b,
                                                 (short)0, acc1, false, false);
  }
}

// ---------------------------------------------------------------------------
// Kernel A: lin = query @ [W_off | W_attn] + bias -> raw logits in workspace.
// grid = 18 N-tiles x 32 M-groups = 576 blocks of 256 threads.
// ---------------------------------------------------------------------------
__global__ __launch_bounds__(256) void k_gemm_logits(
    const float* __restrict__ query,
    const float* __restrict__ W_off,  const float* __restrict__ b_off,
    const float* __restrict__ W_attn, const float* __restrict__ b_attn,
    float* __restrict__ offbuf, float* __restrict__ attnbuf) {
  __shared__ v2f Bs[(D_ / 2) * 16];   // 16 KB
  const int lane   = threadIdx.x & 31;
  const int w      = threadIdx.x >> 5;
  const int nTile  = blockIdx.x % 18;
  const int mGroup = blockIdx.x / 18;

  const float* Bm;
  const float* bias;
  float* Dst;
  int ldb, col0;
  if (nTile < 12) { Bm = W_off;  bias = b_off;  Dst = offbuf;  ldb = LOFF_; col0 = nTile * 16; }
  else            { Bm = W_attn; bias = b_attn; Dst = attnbuf; ldb = LATT_; col0 = (nTile - 12) * 16; }

  stage_B_lds(Bs, Bm, ldb, col0, threadIdx.x);
  __syncthreads();

  const int mTile0 = mGroup * 16 + w * 2;
  const int row0   = mTile0 * 16;
  const float* A0  = query + (size_t)row0 * D_;
  const float* A1  = A0 + 16 * D_;

  v8f acc0 = {0.f,0.f,0.f,0.f,0.f,0.f,0.f,0.f};
  v8f acc1 = {0.f,0.f,0.f,0.f,0.f,0.f,0.f,0.f};
  wmma_dual_tile(A0, A1, Bs, lane, acc0, acc1);

  const int n  = lane & 15;
  const float bv = bias[col0 + n];
#pragma unroll
  for (int r = 0; r < 8; ++r) {
    const int m0 = row0 + r + ((lane >> 4) << 3);
    Dst[(size_t)m0 * ldb + col0 + n]        = acc0[r] + bv;
    Dst[(size_t)(m0 + 16) * ldb + col0 + n] = acc1[r] + bv;
  }
}

// ---------------------------------------------------------------------------
// Kernel B: deformable bilinear sampling + attention combine.
// grid = 8192 blocks (one per (b,q)); block = 256 (wave = head, lane = channel).
// ---------------------------------------------------------------------------
__device__ __forceinline__ float ms_corner(const float* __restrict__ vp, int Wd, int Hd,
                                           int x, int y, float w) {
  if (x < 0 || x >= Wd || y < 0 || y >= Hd) return 0.f;   // wave-uniform branch
  return w * vp[y * Wd + x];
}

__global__ __launch_bounds__(256) void k_sample(
    const float* __restrict__ refpts,
    const float* __restrict__ v0, const float* __restrict__ v1, const float* __restrict__ v2,
    const float* __restrict__ offbuf, const float* __restrict__ attnbuf,
    float* __restrict__ tmp) {
  const int bq = blockIdx.x;           // b*Q + q
  const int b  = bq >> 11;             // Q_ = 2048
  const int h  = threadIdx.x >> 5;
  const int c  = threadIdx.x & 31;

  const float refx = refpts[bq * 2 + 0];
  const float refy = refpts[bq * 2 + 1];
  const float* offl  = offbuf  + (size_t)bq * LOFF_ + h * (NL_ * NPT_ * 2);
  const float* attnl = attnbuf + (size_t)bq * LATT_ + h * (NL_ * NPT_);

  const float* vbase[3] = { v0, v1, v2 };
  const int    dims[3]  = { 128, 64, 32 };

  float out_c = 0.f;
#pragma unroll
  for (int l = 0; l < NL_; ++l) {
    const int Wd = dims[l];
    const int Hd = dims[l];
    const float* vp = vbase[l] + (size_t)(b * D_ + h * HD_ + c) * Hd * Wd;

    // softmax over the 4 points of this (head, level)
    const float lg0 = attnl[l * 4 + 0];
    const float lg1 = attnl[l * 4 + 1];
    const float lg2 = attnl[l * 4 + 2];
    const float lg3 = attnl[l * 4 + 3];
    const float mx = fmaxf(fmaxf(lg0, lg1), fmaxf(lg2, lg3));
    const float e0 = __expf(lg0 - mx);
    const float e1 = __expf(lg1 - mx);
    const float e2 = __expf(lg2 - mx);
    const float e3 = __expf(lg3 - mx);
    const float inv = 1.f / (e0 + e1 + e2 + e3);
    const float aw[4] = { e0 * inv, e1 * inv, e2 * inv, e3 * inv };

#pragma unroll
    for (int p = 0; p < NPT_; ++p) {
      const float ox = tanhf(offl[l * 8 + p * 2 + 0]) * 0.5f;
      const float oy = tanhf(offl[l * 8 + p * 2 + 1]) * 0.5f;
      // grid = (ref+off)*2-1 ; ix = ((gx+1)*W-1)*0.5 = (ref+off)*W - 0.5
      const float ix = (refx + ox) * (float)Wd - 0.5f;
      const float iy = (refy + oy) * (float)Hd - 0.5f;
      const float x0f = floorf(ix), y0f = floorf(iy);
      const int   x0  = (int)x0f,   y0  = (int)y0f;
      const float wx1 = ix - x0f,   wy1 = iy - y0f;
      const float wx0 = 1.f - wx1,  wy0 = 1.f - wy1;
      float s;
      s  = ms_corner(vp, Wd, Hd, x0,     y0,     wx0 * wy0);
      s += ms_corner(vp, Wd, Hd, x0 + 1, y0,     wx1 * wy0);
      s += ms_corner(vp, Wd, Hd, x0,     y0 + 1, wx0 * wy1);
      s += ms_corner(vp, Wd, Hd, x0 + 1, y0 + 1, wx1 * wy1);
      out_c = fmaf(aw[p], s, out_c);
    }
  }
  tmp[(size_t)bq * D_ + h * HD_ + c] = out_c;
}

// ---------------------------------------------------------------------------
// Kernel C: out = tmp @ W_out + b_out.
// grid = 16 N-tiles x 32 M-groups = 512 blocks of 256 threads.
// ---------------------------------------------------------------------------
__global__ __launch_bounds__(256) void k_gemm_out(
    const float* __restrict__ tmp,
    const float* __restrict__ W_out, const float* __restrict__ b_out,
    float* __restrict__ out) {
  __shared__ v2f Bs[(D_ / 2) * 16];   // 16 KB
  const int lane   = threadIdx.x & 31;
  const int w      = threadIdx.x >> 5;
  const int nTile  = blockIdx.x & 15;
  const int mGroup = blockIdx.x >> 4;
  const int col0   = nTile * 16;

  stage_B_lds(Bs, W_out, D_, col0, threadIdx.x);
  __syncthreads();

  const int mTile0 = mGroup * 16 + w * 2;
  const int row0   = mTile0 * 16;
  const float* A0  = tmp + (size_t)row0 * D_;
  const float* A1  = A0 + 16 * D_;

  v8f acc0 = {0.f,0.f,0.f,0.f,0.f,0.f,0.f,0.f};
  v8f acc1 = {0.f,0.f,0.f,0.f,0.f,0.f,0.f,0.f};
  wmma_dual_tile(A0, A1, Bs, lane, acc0, acc1);

  const int n  = lane & 15;
  const float bv = b_out[col0 + n];
#pragma unroll
  for (int r = 0; r < 8; ++r) {
    const int m0 = row0 + r + ((lane >> 4) << 3);
    out[(size_t)m0 * D_ + col0 + n]        = acc0[r] + bv;
    out[(size_t)(m0 + 16) * D_ + col0 + n] = acc1[r] + bv;
  }
}

// ---------------------------------------------------------------------------
extern "C" void kernel_launch(void* const* d_in, const int* in_sizes, int n_in,
                              void* d_out, int out_size, void* d_ws, size_t ws_size,
                              hipStream_t stream) {
  const float* query  = (const float*)d_in[0];
  const float* refp   = (const float*)d_in[1];
  const float* v0     = (const float*)d_in[2];
  const float* v1     = (const float*)d_in[3];
  const float* v2     = (const float*)d_in[4];
  const float* W_off  = (const float*)d_in[5];
  const float* b_off  = (const float*)d_in[6];
  const float* W_attn = (const float*)d_in[7];
  const float* b_attn = (const float*)d_in[8];
  const float* W_out  = (const float*)d_in[9];
  const float* b_out  = (const float*)d_in[10];
  float* out = (float*)d_out;

  // workspace: offbuf (8192x192) | attnbuf (8192x96) | tmp (8192x256)  ~17 MB
  float* offbuf  = (float*)d_ws;
  float* attnbuf = offbuf + (size_t)ROWS_ * LOFF_;
  float* tmpbuf  = attnbuf + (size_t)ROWS_ * LATT_;

  k_gemm_logits<<<18 * MGRP_, 256, 0, stream>>>(query, W_off, b_off,
                                                W_attn, b_attn, offbuf, attnbuf);
  k_sample<<<ROWS_, 256, 0, stream>>>(refp, v0, v1, v2, offbuf, attnbuf, tmpbuf);
  k_gemm_out<<<16 * MGRP_, 256, 0, stream>>>(tmpbuf, W_out, b_out, out);
}